// CenterNet_64965675319610
// MI455X (gfx1250) — compile-verified
//
#include <hip/hip_runtime.h>
#include <stdint.h>

// CenterNet decode for B=16, C=80, H=W=128, K=100, NMS=3, DOWN_RATIO=4.
// d_in[0]=hm [16,80,128,128] f32, d_in[1]=wh [16,2,128,128] f32,
// d_in[2]=reg [16,2,128,128] f32.
// d_out = boxes(16*100*4) ++ scores(16*100) ++ classes(16*100) = 9600 f32.

#define CN_B 16
#define CN_C 80
#define CN_HW 16384   // 128*128
#define CN_K 100
#define PCAP 4096     // max strict 3x3 local maxima in 128x128 = 64*64
#define NSORT2 8192   // next pow2 >= 80*100
#define EQCAP 256

__device__ __forceinline__ float clamp_logit(float v) {
    // logit(1e-4)..logit(1-1e-4); clip(sigmoid(x)) == sigmoid(clamp(x)),
    // monotone => sorting on clamped logits == sorting on clipped scores.
    const float LO = -9.2102403669758f;
    const float HI =  9.2102403669758f;
    return fminf(fmaxf(v, LO), HI);
}

__device__ __forceinline__ unsigned order_f32(float f) {
    unsigned b = __float_as_uint(f);
    return b ^ ((b & 0x80000000u) ? 0xFFFFFFFFu : 0x80000000u);
}

// Descending bitonic sort of n (power of two) u64 keys in LDS.
__device__ __forceinline__ void bitonic_desc(unsigned long long* buf, int n,
                                             int tid, int nthr) {
    for (int k = 2; k <= n; k <<= 1) {
        for (int j = k >> 1; j > 0; j >>= 1) {
            __syncthreads();
            for (int i = tid; i < n; i += nthr) {
                int ixj = i ^ j;
                if (ixj > i) {
                    unsigned long long a = buf[i];
                    unsigned long long b = buf[ixj];
                    bool up = ((i & k) == 0);           // descending block
                    if (up ? (a < b) : (a > b)) { buf[i] = b; buf[ixj] = a; }
                }
            }
        }
    }
    __syncthreads();
}

// ---------------------------------------------------------------------------
// Kernel 1: one block per (b,c) plane. Single-pass rolling-register 3x3 NMS
// (each hm element loaded exactly once), peak compaction into LDS, then an
// exact top-100 via 4-pass MSD radix SELECT (not a sort) + compaction.
// Candidate u64 = (ordered(clamped_logit) << 21) | (0x1FFFFF - ((c<<14)|idx))
// so descending order == (score desc, class asc, idx asc), matching
// jax.lax.top_k's lower-index-first tie rule at both levels.
// ---------------------------------------------------------------------------
__global__ __launch_bounds__(256) void
cn_nms_select_kernel(const float* __restrict__ hm,
                     unsigned long long* __restrict__ cand) {
    __shared__ unsigned long long buf[PCAP];     // 32 KB peak store
    __shared__ float vrow[2][2][128];            // double-buffered col-max row
    __shared__ unsigned pmask[512];              // peak bitmap (fallback only)
    __shared__ unsigned long long eq[EQCAP];     // boundary-key candidates
    __shared__ int hist[256];
    __shared__ int s_np, s_cnt, s_ecnt, s_need;
    __shared__ unsigned s_prefix;

    const int tid = threadIdx.x;
    const int bc  = blockIdx.x;                  // b*80 + c
    const int c   = bc % CN_C;
    const float* plane = hm + (size_t)bc * CN_HW;

    for (int i = tid; i < 512; i += 256) pmask[i] = 0u;
    if (tid == 0) s_np = 0;

    const int half = tid >> 7;                   // 0: rows 0..63, 1: 64..127
    const int x    = tid & 127;
    const int y0   = half << 6;
    const float NEG = -1e30f;

    float r0 = (y0 == 0) ? NEG : clamp_logit(plane[(y0 - 1) * 128 + x]);
    float r1 = clamp_logit(plane[y0 * 128 + x]);
    __syncthreads();

    for (int i = 0; i < 64; ++i) {
        const int y = y0 + i;
        // gfx1250 prefetch: one lane per wave pulls the cacheline 8 rows ahead
        if (((tid & 31) == 0) && (y + 8 < 128))
            __builtin_prefetch(plane + (y + 8) * 128 + x, 0, 3);
        float r2 = (y < 127) ? clamp_logit(plane[(y + 1) * 128 + x]) : NEG;
        float vm = fmaxf(r0, fmaxf(r1, r2));     // vertical 3-max at (y, x)
        const int p = i & 1;                     // double buffer: 1 barrier/row
        vrow[half][p][x] = vm;
        __syncthreads();
        float lm = (x > 0)   ? vrow[half][p][x - 1] : NEG;
        float rm = (x < 127) ? vrow[half][p][x + 1] : NEG;
        float hmax = fmaxf(vm, fmaxf(lm, rm));
        if (r1 >= hmax) {                        // r1 <= hmax => equality test
            unsigned key = order_f32(r1);
            unsigned idx = (unsigned)(y * 128 + x);
            unsigned payload = ((unsigned)c << 14) | idx;
            unsigned long long cd = ((unsigned long long)key << 21)
                                  | (unsigned long long)(0x1FFFFFu - payload);
            int slot = atomicAdd(&s_np, 1);
            if (slot < PCAP) buf[slot] = cd;
            atomicOr(&pmask[idx >> 5], 1u << (idx & 31));
        }
        r0 = r1; r1 = r2;
    }
    __syncthreads();

    // Fallback (<100 peaks, measure-zero for iid data): reference fills with
    // zero-score entries at the lowest non-peak flat indices. Sentinel key=1
    // sorts below every real peak key (~1e9+) and above empty slots.
    int np = s_np;
    if (np < CN_K) {                             // uniform branch (np shared)
        unsigned idx = (unsigned)tid;            // 256 lowest indices: plenty
        if (!((pmask[idx >> 5] >> (idx & 31)) & 1u)) {
            unsigned payload = ((unsigned)c << 14) | idx;
            unsigned long long cd = (1ull << 21)
                                  | (unsigned long long)(0x1FFFFFu - payload);
            int slot = atomicAdd(&s_np, 1);
            if (slot < PCAP) buf[slot] = cd;
        }
    }
    __syncthreads();
    np = min(s_np, PCAP);                        // >= 100 guaranteed

    // ---- exact 100th-largest key via MSD radix select (4 byte passes) ----
    if (tid == 0) { s_prefix = 0u; s_need = CN_K; }
    for (int lvl = 3; lvl >= 0; --lvl) {
        for (int i = tid; i < 256; i += 256) hist[i] = 0;
        __syncthreads();
        unsigned pfx = s_prefix;
        for (int i = tid; i < np; i += 256) {
            unsigned key = (unsigned)(buf[i] >> 21);
            bool m = (lvl == 3) || ((key >> (8 * (lvl + 1))) == pfx);
            if (m) atomicAdd(&hist[(key >> (8 * lvl)) & 0xFF], 1);
        }
        __syncthreads();
        if (tid == 0) {
            int acc = 0, need = s_need, bin = 255;
            for (; bin > 0; --bin) {
                int h = hist[bin];
                if (acc + h >= need) break;
                acc += h;
            }
            s_need   = need - acc;               // equals still required
            s_prefix = (pfx << 8) | (unsigned)bin;
        }
        __syncthreads();
    }
    const unsigned T = s_prefix;                 // 100th-largest key value
    const int need   = s_need;                   // # key==T entries to keep
    if (tid == 0) { s_cnt = 0; s_ecnt = 0; }
    __syncthreads();

    // ---- compact: strict winners directly out, boundary ties collected ----
    unsigned long long* outp = cand + (size_t)bc * CN_K;
    for (int i = tid; i < np; i += 256) {
        unsigned long long cd = buf[i];
        unsigned key = (unsigned)(cd >> 21);
        if (key > T) {
            int slot = atomicAdd(&s_cnt, 1);     // slot < 100 - need < 100
            outp[slot] = cd;
        } else if (key == T) {
            int e = atomicAdd(&s_ecnt, 1);
            if (e < EQCAP) eq[e] = cd;
        }
    }
    __syncthreads();
    if (tid == 0) {
        int G    = s_cnt;                        // == CN_K - need
        int ecnt = min(s_ecnt, EQCAP);           // ecnt >= need always
        for (int j = 0; j < need; ++j) {         // pick largest u64 ties:
            int best = 0; unsigned long long bv = 0ull;   // == lowest idx
            for (int e = 0; e < ecnt; ++e) {
                unsigned long long v = eq[e];
                if (v > bv) { bv = v; best = e; }
            }
            outp[G + j] = bv;
            eq[best] = 0ull;                     // real cds >= 1<<21 > 0
        }
    }
}

// ---------------------------------------------------------------------------
// Kernel 2: one block per batch. Async-copy the 8000 candidates to LDS
// (gfx1250 GLOBAL_LOAD_ASYNC_TO_LDS_B64 + s_wait_asynccnt), bitonic sort,
// decode top-100, gather reg/wh, emit boxes*DOWN_RATIO, scores, classes.
// ---------------------------------------------------------------------------
__global__ __launch_bounds__(256) void
cn_decode_kernel(const unsigned long long* __restrict__ cand,
                 const float* __restrict__ wh,
                 const float* __restrict__ reg,
                 float* __restrict__ out) {
    __shared__ unsigned long long buf[NSORT2];   // 64 KB
    const int tid = threadIdx.x;
    const int b   = blockIdx.x;
    const unsigned long long* src = cand + (size_t)b * (CN_C * CN_K);

    for (int i = tid; i < CN_C * CN_K; i += 256) {
        unsigned lds_addr = (unsigned)(uintptr_t)&buf[i]; // low 32b = LDS addr
        const unsigned long long* gp = src + i;
        asm volatile("global_load_async_to_lds_b64 %0, %1, off"
                     :: "v"(lds_addr), "v"(gp) : "memory");
    }
    for (int i = CN_C * CN_K + tid; i < NSORT2; i += 256) buf[i] = 0ull;
    asm volatile("s_wait_asynccnt 0x0" ::: "memory");
    __syncthreads();

    bitonic_desc(buf, NSORT2, tid, 256);

    if (tid < CN_K) {
        unsigned long long v = buf[tid];
        unsigned key     = (unsigned)(v >> 21);
        unsigned payload = 0x1FFFFFu - ((unsigned)v & 0x1FFFFFu);
        int cls = (int)(payload >> 14);
        int idx = (int)(payload & 0x3FFFu);
        float fx = (float)(idx & 127);
        float fy = (float)(idx >> 7);

        float score;
        if (key <= 1u) {
            score = 0.0f;                        // suppressed-fill sentinel
        } else {
            unsigned bits = (key & 0x80000000u) ? (key ^ 0x80000000u) : ~key;
            float lg = __uint_as_float(bits);
            float s  = 1.0f / (1.0f + expf(-lg));
            score = fminf(fmaxf(s, 1e-4f), 1.0f - 1e-4f);
        }

        const float* regb = reg + (size_t)b * 2 * CN_HW;
        const float* whb  = wh  + (size_t)b * 2 * CN_HW;
        float xs = fx + regb[idx];
        float ys = fy + regb[CN_HW + idx];
        float ww = whb[idx];
        float hh = whb[CN_HW + idx];

        const int o = b * CN_K + tid;
        out[(size_t)o * 4 + 0] = 4.0f * (xs - ww * 0.5f);
        out[(size_t)o * 4 + 1] = 4.0f * (ys - hh * 0.5f);
        out[(size_t)o * 4 + 2] = 4.0f * (xs + ww * 0.5f);
        out[(size_t)o * 4 + 3] = 4.0f * (ys + hh * 0.5f);
        out[CN_B * CN_K * 4 + o] = score;        // scores at offset 6400
        out[CN_B * CN_K * 5 + o] = (float)cls;   // classes at offset 8000
    }
}

extern "C" void kernel_launch(void* const* d_in, const int* in_sizes, int n_in,
                              void* d_out, int out_size, void* d_ws, size_t ws_size,
                              hipStream_t stream) {
    (void)in_sizes; (void)n_in; (void)out_size; (void)ws_size;
    const float* hm  = (const float*)d_in[0];
    const float* wh  = (const float*)d_in[1];
    const float* reg = (const float*)d_in[2];
    unsigned long long* cand = (unsigned long long*)d_ws;  // 16*80*100*8 = 1 MB

    cn_nms_select_kernel<<<dim3(CN_B * CN_C), dim3(256), 0, stream>>>(hm, cand);
    cn_decode_kernel<<<dim3(CN_B), dim3(256), 0, stream>>>(cand, wh, reg,
                                                           (float*)d_out);
}